// WeightedLossModel_34694745817489
// MI455X (gfx1250) — compile-verified
//
#include <hip/hip_runtime.h>
#include <math.h>
#include <stdint.h>

typedef float v2f __attribute__((ext_vector_type(2)));
typedef float v8f __attribute__((ext_vector_type(8)));
typedef unsigned int u32x4 __attribute__((ext_vector_type(4)));
typedef int i32x8 __attribute__((ext_vector_type(8)));
typedef int i32x4 __attribute__((ext_vector_type(4)));

#if __has_builtin(__builtin_amdgcn_tensor_load_to_lds)
#define HAVE_TDM 1
#endif

#define BATCH   512
#define HH      96
#define WW      96
#define PIX     (HH * WW)          // 9216
#define KSZ     25
#define KR      12
#define KELEM   (KSZ * KSZ)        // 625
#define NTOT    (BATCH * PIX)      // 4718592
#define PI_F    3.14159265358979323846f

// per-sample state slots (SoA, each BATCH floats)
#define S_AX    0
#define S_AY    1
#define S_AXY   2
#define S_MUX   3
#define S_MUY   4
#define S_PSXX  5
#define S_PSYY  6
#define S_CSXX  7
#define S_CSYY  8
#define S_BACK  9
#define S_E1    10
#define S_E2    11
#define S_TOT   12
#define NSTATE  13

__device__ __forceinline__ float clampf(float v, float lo, float hi) {
    return fminf(fmaxf(v, lo), hi);
}

// ---------------------------------------------------------------------------
// Init: per-sample image total + state init + control flags
// ---------------------------------------------------------------------------
__global__ void __launch_bounds__(256)
ellip_init(const float* __restrict__ img, float* __restrict__ st, int* __restrict__ ctrl) {
    const int s = blockIdx.x;
    const int tid = threadIdx.x;
    __shared__ float red[256];
    float sum = 0.0f;
    for (int i = tid; i < PIX; i += 256) sum += img[s * PIX + i];
    red[tid] = sum;
    __syncthreads();
    for (int off = 128; off > 0; off >>= 1) {
        if (tid < off) red[tid] += red[tid + off];
        __syncthreads();
    }
    if (tid == 0) {
        st[S_TOT  * BATCH + s] = red[0];
        st[S_AX   * BATCH + s] = 3.0f;
        st[S_AY   * BATCH + s] = 3.0f;
        st[S_AXY  * BATCH + s] = 0.0f;
        st[S_MUX  * BATCH + s] = 0.0f;
        st[S_MUY  * BATCH + s] = 0.0f;
        st[S_PSXX * BATCH + s] = 9999.0f;
        st[S_PSYY * BATCH + s] = 9999.0f;
        st[S_CSXX * BATCH + s] = 0.0f;
        st[S_CSYY * BATCH + s] = 0.0f;
        st[S_BACK * BATCH + s] = 0.0f;
        st[S_E1   * BATCH + s] = 0.0f;
        st[S_E2   * BATCH + s] = 0.0f;
        if (s == 0) { ctrl[0] = 1; ctrl[1] = 0; }   // run=1, iter=0
    }
}

// ---------------------------------------------------------------------------
// Per-iteration moment sums via V_WMMA_F32_16X16X4_F32.
// One workgroup (8 waves) handles 16 samples; M = sample, N = moment,
// K = pixels (4 per WMMA). Columns: 0:Σwk 1:ΣXwk 2:ΣYwk 3:ΣXYwk 4:ΣXXwk 5:ΣYYwk.
// t7 = Σk² accumulated per-lane (each lane owns disjoint (sample,pixel) slots).
// Branchless B-operand: per-lane 0/1 polynomial coefficients chosen once.
// ---------------------------------------------------------------------------
__global__ void __launch_bounds__(256)
ellip_moments(const float* __restrict__ img, const float* __restrict__ st,
              float* __restrict__ ts, const int* __restrict__ ctrl) {
    if (ctrl[0] == 0) return;
    const int tid   = threadIdx.x;
    const int wave  = tid >> 5;
    const int lane  = tid & 31;
    const int mrow  = lane & 15;       // sample-in-group / moment column
    const int khalf = lane >> 4;       // 0: K=0,1   1: K=2,3
    const int s0    = blockIdx.x * 16;
    const int smp   = s0 + mrow;

    const float ax   = st[S_AX  * BATCH + smp];
    const float ay   = st[S_AY  * BATCH + smp];
    const float axy  = st[S_AXY * BATCH + smp];
    const float mux  = st[S_MUX * BATCH + smp];
    const float muy  = st[S_MUY * BATCH + smp];
    const float back = st[S_BACK* BATCH + smp];

    float rho = clampf(axy / (ax * ay + 1e-10f), -0.99f, 0.99f);
    const float arb  = fmaxf(2.0f * (1.0f - rho * rho), 1e-10f);
    const float Aamp = 1.0f / (2.0f * PI_F * ax * ay * sqrtf(1.0f - rho * rho));
    const float ia   = 1.0f / (arb * ax * ax);
    const float ib   = 1.0f / (arb * ay * ay);
    const float cc   = 2.0f * axy / (arb * ax * ax * ay * ay);

    // per-lane moment-polynomial coefficients (selected once, branchless loop)
    const float c1  = (mrow == 0) ? 1.0f : 0.0f;
    const float cx  = (mrow == 1) ? 1.0f : 0.0f;
    const float cy  = (mrow == 2) ? 1.0f : 0.0f;
    const float cxy = (mrow == 3) ? 1.0f : 0.0f;
    const float cxx = (mrow == 4) ? 1.0f : 0.0f;
    const float cyy = (mrow == 5) ? 1.0f : 0.0f;

    v8f acc = {0.0f, 0.0f, 0.0f, 0.0f, 0.0f, 0.0f, 0.0f, 0.0f};
    float t7 = 0.0f;

    const float2* __restrict__ img2 = (const float2*)(img + (size_t)smp * PIX);
    int p = wave * 4 + khalf * 2;      // this lane's first pixel; step 32/iter
    int x = p, y = 0;

#pragma unroll 4
    for (int it = 0; it < PIX / 32; ++it) {      // 288 iterations per wave
        const float Yv    = (float)y - 47.5f;
        const float Xv0   = (float)x - 47.5f;
        const float Xv1   = Xv0 + 1.0f;
        const float Yc    = Yv - muy;
        const float yterm = Yc * Yc * ib;
        const float ccY   = cc * Yc;
        const float bbase = c1 + cy * Yv + cyy * (Yv * Yv);
        const float bx    = cx + cxy * Yv;
        const float2 iv   = img2[p >> 1];        // pixels p, p+1 (b64 load)

        const float Xc0 = Xv0 - mux;
        float et0 = Xc0 * Xc0 * ia + yterm - ccY * Xc0;
        et0 = clampf(et0, 0.0f, 50.0f);
        const float k0  = Aamp * __expf(-et0);
        const float wk0 = (iv.x - back) * k0;
        const float bj0 = bbase + Xv0 * bx + cxx * (Xv0 * Xv0);

        const float Xc1 = Xv1 - mux;
        float et1 = Xc1 * Xc1 * ia + yterm - ccY * Xc1;
        et1 = clampf(et1, 0.0f, 50.0f);
        const float k1  = Aamp * __expf(-et1);
        const float wk1 = (iv.y - back) * k1;
        const float bj1 = bbase + Xv1 * bx + cxx * (Xv1 * Xv1);

        t7 += k0 * k0 + k1 * k1;
        v2f av = {wk0, wk1};
        v2f bv = {bj0, bj1};
        acc = __builtin_amdgcn_wmma_f32_16x16x4_f32(false, av, false, bv,
                                                    (short)0, acc, false, false);
        p += 32;
        x += 32;
        if (x >= WW) { x -= WW; y += 1; }        // cndmask, no branch
    }

    __shared__ float red[8][16][6];
    __shared__ float t7l[8][32];
    if (mrow < 6) {
#pragma unroll
        for (int r = 0; r < 8; ++r)
            red[wave][r + 8 * khalf][mrow] = acc[r];   // D: vgpr r ↦ M=r(+8), N=lane%16
    }
    t7l[wave][lane] = t7;
    __syncthreads();

    if (tid < 112) {                                   // 16 samples × 7 sums
        const int m = tid / 7;
        const int j = tid - m * 7;
        float s = 0.0f;
        if (j < 6) {
            for (int w = 0; w < 8; ++w) s += red[w][m][j];
        } else {
            for (int w = 0; w < 8; ++w) s += t7l[w][m] + t7l[w][m + 16];
        }
        ts[j * BATCH + s0 + m] = s;
    }
}

// ---------------------------------------------------------------------------
// Per-iteration state update + global convergence test (matches while_loop).
// ---------------------------------------------------------------------------
__global__ void __launch_bounds__(512)
ellip_update(float* __restrict__ st, const float* __restrict__ ts, int* __restrict__ ctrl) {
    if (ctrl[0] == 0) return;
    const int s = threadIdx.x;
    const float t2 = ts[0 * BATCH + s];
    const float t3 = ts[1 * BATCH + s];
    const float t4 = ts[2 * BATCH + s];
    const float t1 = ts[3 * BATCH + s];
    const float t5 = ts[4 * BATCH + s];
    const float t6 = ts[5 * BATCH + s];
    const float t7 = ts[6 * BATCH + s];
    const float total = st[S_TOT * BATCH + s];

    const float flux  = t2 / (t7 + 1e-10f);
    const float nback = (total - flux) / (float)PIX;
    const float t2s   = fmaxf(t2, 1e-10f);
    const float nmux  = t3 / t2s;
    const float nmuy  = t4 / t2s;
    const float sigxx = t5 / t2s - (t3 / t2s) * (t3 / t2s);
    const float sigyy = t6 / t2s - (t4 / t2s) * (t4 / t2s);
    const float sigxy = t1 / t2s - t3 * t4 / (t2s * t2s);
    const float nax   = sqrtf(clampf(sigxx * 2.0f, 0.81f, 100.0f));
    const float nay   = sqrtf(clampf(sigyy * 2.0f, 0.81f, 100.0f));
    const float naxy  = 2.0f * sigxy;
    const float den   = sigxx + sigyy + 1e-10f;
    const float e1    = clampf((sigxx - sigyy) / den, -0.99f, 0.99f);
    const float e2    = clampf(2.0f * sigxy / den, -0.99f, 0.99f);

    const float psxx = st[S_CSXX * BATCH + s];  // old curr becomes prev
    const float psyy = st[S_CSYY * BATCH + s];

    st[S_AX   * BATCH + s] = nax;
    st[S_AY   * BATCH + s] = nay;
    st[S_AXY  * BATCH + s] = naxy;
    st[S_MUX  * BATCH + s] = nmux;
    st[S_MUY  * BATCH + s] = nmuy;
    st[S_PSXX * BATCH + s] = psxx;
    st[S_PSYY * BATCH + s] = psyy;
    st[S_CSXX * BATCH + s] = sigxx;
    st[S_CSYY * BATCH + s] = sigyy;
    st[S_BACK * BATCH + s] = nback;
    st[S_E1   * BATCH + s] = e1;
    st[S_E2   * BATCH + s] = e2;

    __shared__ float red[512];
    red[s] = fmaxf(fabsf(sigxx - psxx), fabsf(sigyy - psyy));
    __syncthreads();
    for (int off = 256; off > 0; off >>= 1) {
        if (s < off) red[s] = fmaxf(red[s], red[s + off]);
        __syncthreads();
    }
    if (s == 0) {
        const int itn = ctrl[1] + 1;
        ctrl[1] = itn;
        const int conv = (red[0] < 1e-3f) ? 1 : 0;
        ctrl[0] = (itn < 100 && !conv) ? 1 : 0;
    }
}

// ---------------------------------------------------------------------------
// Per-sample kernel sum (for normalization).
// ---------------------------------------------------------------------------
__global__ void __launch_bounds__(256)
ksum_kernel(const float* __restrict__ ker, float* __restrict__ ks) {
    const int s = blockIdx.x;
    const int tid = threadIdx.x;
    __shared__ float red[256];
    float sum = 0.0f;
    for (int i = tid; i < KELEM; i += 256) sum += ker[s * KELEM + i];
    red[tid] = sum;
    __syncthreads();
    for (int off = 128; off > 0; off >>= 1) {
        if (tid < off) red[tid] += red[tid + off];
        __syncthreads();
    }
    if (tid == 0) ks[s] = red[0];
}

// ---------------------------------------------------------------------------
// Exact per-sample order statistic via 32-pass radix binary search.
// ---------------------------------------------------------------------------
__device__ float radix_select(const unsigned* m, int k, int tid, int* red) {
    unsigned prefix = 0u;
    for (int bit = 31; bit >= 0; --bit) {
        const unsigned trial = prefix | (1u << bit);
        int c = 0;
#pragma unroll
        for (int t = 0; t < 36; ++t) c += (m[t] < trial) ? 1 : 0;
        red[tid] = c;
        __syncthreads();
        for (int off = 128; off > 0; off >>= 1) {
            if (tid < off) red[tid] += red[tid + off];
            __syncthreads();
        }
        const int tot = red[0];
        __syncthreads();
        if (tot <= k) prefix = trial;
    }
    const unsigned u = (prefix & 0x80000000u) ? (prefix ^ 0x80000000u) : ~prefix;
    return __uint_as_float(u);
}

__global__ void __launch_bounds__(256)
pct_kernel(const float* __restrict__ blurred, float* __restrict__ vmn, float* __restrict__ vmx) {
    const int s = blockIdx.x;
    const int tid = threadIdx.x;
    __shared__ int red[256];
    unsigned m[36];
#pragma unroll
    for (int t = 0; t < 36; ++t) {
        const unsigned u = __float_as_uint(blurred[s * PIX + tid + t * 256]);
        m[t] = (u & 0x80000000u) ? ~u : (u | 0x80000000u);
    }
    const float lo = radix_select(m, 92, tid, red);    // round(0.01*9215)
    const float hi = radix_select(m, 9123, tid, red);  // round(0.99*9215)
    if (tid == 0) { vmn[s] = lo; vmx[s] = hi; }
}

// ---------------------------------------------------------------------------
// Fused depthwise 25x25 SAME conv + percentile normalize/clip + weighted MSE.
// Image tile staged into LDS by the Tensor Data Mover when available.
// ---------------------------------------------------------------------------
__global__ void __launch_bounds__(256)
reblur_kernel(const float* __restrict__ ypred, const float* __restrict__ ker,
              const float* __restrict__ blurred, const float* __restrict__ wmap,
              const float* __restrict__ ks, const float* __restrict__ vmn,
              const float* __restrict__ vmx, float* __restrict__ partial) {
    const int s = blockIdx.x;
    const int tid = threadIdx.x;
    __shared__ float simg[PIX];
    __shared__ float sker[KELEM];
    __shared__ float red[256];

#ifdef HAVE_TDM
    if (tid < 32) {
        // D# group0: count=1 | lds_addr | global_addr[56:0] | type=2
        const unsigned lds_off = (unsigned)(uintptr_t)(void*)simg;
        const unsigned long long ga =
            (unsigned long long)(uintptr_t)(ypred + (size_t)s * PIX);
        u32x4 g0;
        g0[0] = 1u;                                       // count=1, user mode
        g0[1] = lds_off;                                  // lds_addr (bytes)
        g0[2] = (unsigned)(ga & 0xffffffffu);             // global_addr lo
        g0[3] = (unsigned)((ga >> 32) & 0x01ffffffu) | (2u << 30); // hi | type=2
        // D# group1: 1-D tensor of 9216 f32, tile 9216
        i32x8 g1;
        g1[0] = 2 << 16;                  // data_size = 4 bytes
        g1[1] = (int)(9216u << 16);       // tensor_dim0[15:0] (bits 63:48)
        g1[2] = (int)(1u << 16);          // tensor_dim0 hi=0 | tensor_dim1=1
        g1[3] = (int)(9216u << 16);       // tensor_dim1 hi=0 | tile_dim0=9216
        g1[4] = 0;                        // tile_dim1=0, tile_dim2=0 (unused)
        g1[5] = 9216;                     // tensor_dim0_stride lo
        g1[6] = 0;
        g1[7] = 0;
        i32x4 gz4 = {0, 0, 0, 0};
        i32x8 gz8 = {0, 0, 0, 0, 0, 0, 0, 0};
        __builtin_amdgcn_tensor_load_to_lds(g0, g1, gz4, gz4, gz8, 0);
        __builtin_amdgcn_s_wait_tensorcnt(0);
    }
#else
    for (int i = tid; i < PIX; i += 256) {
        simg[i] = ypred[s * PIX + i];
        __builtin_prefetch(&ypred[s * PIX + i + 2048], 0, 1);
    }
#endif
    for (int i = tid; i < KELEM; i += 256) sker[i] = ker[s * KELEM + i];
    __syncthreads();

    const float invk   = 1.0f / (ks[s] + 1e-6f);
    const float vmin   = vmn[s];
    const float iscale = 1.0f / (vmx[s] - vmin + 1e-6f);

    float lsum = 0.0f;
    for (int idx = tid; idx < PIX; idx += 256) {
        const int y = idx / WW;
        const int x = idx - y * WW;
        float acc = 0.0f;
        for (int dy = 0; dy < KSZ; ++dy) {
            const int iy = y + dy - KR;
            if ((unsigned)iy < (unsigned)HH) {
                const float* ir = &simg[iy * WW];
                const float* kr = &sker[dy * KSZ];
                for (int dx = 0; dx < KSZ; ++dx) {
                    const int ix = x + dx - KR;
                    if ((unsigned)ix < (unsigned)WW) acc += ir[ix] * kr[dx];
                }
            }
        }
        float r = (acc * invk - vmin) * iscale;
        r = clampf(r, 0.0f, 2.0f);
        const float d = r - blurred[s * PIX + idx];
        lsum += d * d * wmap[s * PIX + idx];
    }
    red[tid] = lsum;
    __syncthreads();
    for (int off = 128; off > 0; off >>= 1) {
        if (tid < off) red[tid] += red[tid + off];
        __syncthreads();
    }
    if (tid == 0) partial[s] = red[0];
}

// ---------------------------------------------------------------------------
// Weighted MSE partial sums.
// ---------------------------------------------------------------------------
__global__ void __launch_bounds__(256)
wmse_kernel(const float* __restrict__ yt, const float* __restrict__ yp,
            const float* __restrict__ w, float* __restrict__ partial) {
    const int tid = threadIdx.x;
    __shared__ float red[256];
    float sum = 0.0f;
    for (int i = blockIdx.x * 256 + tid; i < NTOT; i += 512 * 256) {
        const float d = yp[i] - yt[i];
        sum += d * d * w[i];
    }
    red[tid] = sum;
    __syncthreads();
    for (int off = 128; off > 0; off >>= 1) {
        if (tid < off) red[tid] += red[tid + off];
        __syncthreads();
    }
    if (tid == 0) partial[blockIdx.x] = red[0];
}

// ---------------------------------------------------------------------------
// Final combine: ellip loss over batch + fixed-order sums of partials.
// ---------------------------------------------------------------------------
__global__ void __launch_bounds__(512)
final_kernel(const float* __restrict__ st0, const float* __restrict__ st1,
             const float* __restrict__ pms, const float* __restrict__ prb,
             float* __restrict__ out) {
    const int s = threadIdx.x;
    float e1t = st0[S_E1 * BATCH + s]; if (e1t != e1t) e1t = 0.0f;
    float e2t = st0[S_E2 * BATCH + s]; if (e2t != e2t) e2t = 0.0f;
    float e1p = st1[S_E1 * BATCH + s]; if (e1p != e1p) e1p = 0.0f;
    float e2p = st1[S_E2 * BATCH + s]; if (e2p != e2p) e2p = 0.0f;
    const float ev = (e1t - e1p) * (e1t - e1p) + (e2t - e2p) * (e2t - e2p);

    __shared__ float red[512];
    float esum = 0.0f, msum = 0.0f, rsum = 0.0f;

    red[s] = ev;
    __syncthreads();
    for (int off = 256; off > 0; off >>= 1) {
        if (s < off) red[s] += red[s + off];
        __syncthreads();
    }
    if (s == 0) esum = red[0];
    __syncthreads();

    red[s] = pms[s];
    __syncthreads();
    for (int off = 256; off > 0; off >>= 1) {
        if (s < off) red[s] += red[s + off];
        __syncthreads();
    }
    if (s == 0) msum = red[0];
    __syncthreads();

    red[s] = prb[s];
    __syncthreads();
    for (int off = 256; off > 0; off >>= 1) {
        if (s < off) red[s] += red[s + off];
        __syncthreads();
    }
    if (s == 0) rsum = red[0];

    if (s == 0) {
        const float mse = msum / (float)NTOT;
        const float reb = rsum / (float)NTOT;
        const float el  = esum / (float)BATCH;
        out[0] = 100.0f * mse + 10.0f * reb;   // total_loss
        out[1] = mse;                          // weighted_mse
        out[2] = el;                           // ellip_loss
        out[3] = reb;                          // reblur_loss
    }
}

// ---------------------------------------------------------------------------
extern "C" void kernel_launch(void* const* d_in, const int* in_sizes, int n_in,
                              void* d_out, int out_size, void* d_ws, size_t ws_size,
                              hipStream_t stream) {
    (void)in_sizes; (void)n_in; (void)out_size; (void)ws_size;
    const float* blurred = (const float*)d_in[0];
    const float* kimg    = (const float*)d_in[1];
    const float* wmap    = (const float*)d_in[2];
    const float* ytrue   = (const float*)d_in[3];
    const float* ypred   = (const float*)d_in[4];
    float* out = (float*)d_out;

    float* wsf  = (float*)d_ws;
    float* st0  = wsf;                              // 13*512
    float* st1  = wsf + NSTATE * BATCH;             // 13*512
    float* ts0  = wsf + 26 * BATCH;                 // 7*512
    float* ts1  = wsf + 33 * BATCH;                 // 7*512
    float* ksum = wsf + 40 * BATCH;
    float* vmn  = wsf + 41 * BATCH;
    float* vmx  = wsf + 42 * BATCH;
    float* prb  = wsf + 43 * BATCH;
    float* pms  = wsf + 44 * BATCH;
    int*   c0   = (int*)(wsf + 45 * BATCH);         // run0, iter0
    int*   c1   = c0 + 2;                           // run1, iter1

    ellip_init<<<BATCH, 256, 0, stream>>>(ytrue, st0, c0);
    ellip_init<<<BATCH, 256, 0, stream>>>(ypred, st1, c1);

    for (int it = 0; it < 100; ++it) {
        ellip_moments<<<BATCH / 16, 256, 0, stream>>>(ytrue, st0, ts0, c0);
        ellip_update<<<1, 512, 0, stream>>>(st0, ts0, c0);
        ellip_moments<<<BATCH / 16, 256, 0, stream>>>(ypred, st1, ts1, c1);
        ellip_update<<<1, 512, 0, stream>>>(st1, ts1, c1);
    }

    ksum_kernel<<<BATCH, 256, 0, stream>>>(kimg, ksum);
    pct_kernel<<<BATCH, 256, 0, stream>>>(blurred, vmn, vmx);
    reblur_kernel<<<BATCH, 256, 0, stream>>>(ypred, kimg, blurred, wmap,
                                             ksum, vmn, vmx, prb);
    wmse_kernel<<<512, 256, 0, stream>>>(ytrue, ypred, wmap, pms);
    final_kernel<<<1, 512, 0, stream>>>(st0, st1, pms, prb, out);
}